// GraphConvolution_4243427689158
// MI455X (gfx1250) — compile-verified
//
#include <hip/hip_runtime.h>

typedef __attribute__((ext_vector_type(2))) float v2f;
typedef __attribute__((ext_vector_type(4))) float f4;
typedef __attribute__((ext_vector_type(8))) float v8f;

#define D_IN  512
#define D_OUT 128
// LDS layout: W element (k,n) lives at wlds[(k>>1)*KP_STRIDE + 2n + (k&1)].
// KP_STRIDE = 2*D_OUT + 32 pad; 288 mod 64 == 32 makes the two lane-halves of a
// wave land on disjoint 32-bank sets -> conflict-free ds_load_b64 B fragments.
#define KP_STRIDE 288
#define WLDS_FLOATS ((D_IN / 2) * KP_STRIDE)   // 73728 floats = 288 KB

// sched_group_barrier masks
#define SG_WMMA   0x0008
#define SG_VMEMRD 0x0020
#define SG_DSRD   0x0100

// ---------------------------------------------------------------------------
// Kernel 1: zero the output accumulator (d_out is poisoned by the harness).
// ---------------------------------------------------------------------------
__global__ void gc_zero_f4(f4* __restrict__ p, int n4) {
    int i = blockIdx.x * blockDim.x + threadIdx.x;
    if (i < n4) {
        f4 z = {};
        p[i] = z;
    }
}

// ---------------------------------------------------------------------------
// Kernel 2: support = x @ W + b  via V_WMMA_F32_16X16X4_F32 (fp32 WMMA).
// Block = 256 threads = 8 waves; each wave owns a 16(M) x 128(N) slab
// (8 x v8f accumulators), K walked in steps of 4 (128 WMMA chains).
// W staged once per block into LDS in K-pair-interleaved padded layout so each
// B fragment is a single aligned, bank-conflict-free ds_load_b64.
// The k-loop is software-pipelined (double-buffered A + B fragments) and the
// per-iteration shape [1 VMEM-rd][4 DS-rd][8 WMMA] is pinned with
// sched_group_barrier so DS latency overlaps the previous step's WMMAs.
//
// Fragment layouts (CDNA5 ISA 7.12.2):
//   A 16x4 f32:  lanes 0-15 hold M=lane, K={0,1}; lanes 16-31 hold K={2,3}
//   B 4x16 f32:  lanes 0-15 hold K-rows {0,1} at N=lane; lanes 16-31 K={2,3}
//   C/D 16x16:   VGPR i -> M=i (lanes 0-15) / M=i+8 (lanes 16-31), N=lane&15
// ---------------------------------------------------------------------------
__global__ __launch_bounds__(256) void gc_gemm_wmma_f32(
    const float* __restrict__ x, const float* __restrict__ W,
    const float* __restrict__ bias, float* __restrict__ support, int M)
{
    extern __shared__ float wlds[];   // WLDS_FLOATS

    const int tid = threadIdx.x;

    // Stage W -> LDS. Coalesced float4 global reads (4 consecutive n at one k),
    // interleaved stride-2 LDS stores (conflict-free across a wave).
    {
        const f4* __restrict__ Wv = (const f4*)W;
        #pragma unroll
        for (int it = 0; it < (D_IN * D_OUT / 4) / 256; ++it) {
            const int i = tid + it * 256;
            const int k = i >> 5;              // W row
            const int n = (i & 31) * 4;        // W col (4-wide)
            const f4 w = Wv[i];
            float* dst = wlds + (k >> 1) * KP_STRIDE + (k & 1) + 2 * n;
            dst[0] = w[0]; dst[2] = w[1]; dst[4] = w[2]; dst[6] = w[3];
        }
    }
    __syncthreads();

    const int wave = tid >> 5;
    const int lane = tid & 31;
    const int m0   = (blockIdx.x * 8 + wave) * 16;
    if (m0 >= M) return;              // whole-wave uniform exit: EXEC stays all-1s

    const int half = lane >> 4;       // 0: K+{0,1}, 1: K+{2,3}
    const int l15  = lane & 15;

    v8f acc[8];
    #pragma unroll
    for (int t = 0; t < 8; ++t) acc[t] = (v8f){};

    // Per-lane A row pointer (row m0+l15, pre-offset by the lane-half K shift).
    const float* __restrict__ xrow = x + (size_t)(m0 + l15) * D_IN + half * 2;
    // Per-lane B base: k-pair (k>>1)+half, column pair 2*l15.
    const float* __restrict__ wp = wlds + half * KP_STRIDE + 2 * l15;

    const int KSTEPS = D_IN / 4;      // 128

    // Pipeline prologue: fragments for step 0.
    v2f a_cur = *(const v2f*)(xrow);
    v2f bf[8];
    #pragma unroll
    for (int t = 0; t < 8; ++t)
        bf[t] = *(const v2f*)(wp + t * 32);

    #pragma unroll 2
    for (int s = 0; s < KSTEPS - 1; ++s) {
        const v2f a_use = a_cur;
        // Prefetch step s+1 while step s computes.
        a_cur = *(const v2f*)(xrow + (s + 1) * 4);                 // global_load_b64
        v2f bn[8];
        #pragma unroll
        for (int t = 0; t < 8; ++t)
            bn[t] = *(const v2f*)(wp + (s + 1) * (2 * KP_STRIDE) + t * 32); // ds_load_b64

        #pragma unroll
        for (int t = 0; t < 8; ++t)
            acc[t] = __builtin_amdgcn_wmma_f32_16x16x4_f32(
                false, a_use, false, bf[t], (short)0, acc[t], false, false);

        #pragma unroll
        for (int t = 0; t < 8; ++t) bf[t] = bn[t];   // renamed away by unroll

        // Pin the pipeline shape: prefetches first, then the WMMA block.
        __builtin_amdgcn_sched_group_barrier(SG_VMEMRD, 1, 0);
        __builtin_amdgcn_sched_group_barrier(SG_DSRD,   4, 0);
        __builtin_amdgcn_sched_group_barrier(SG_WMMA,   8, 0);
    }

    // Pipeline epilogue: last step, no prefetch.
    #pragma unroll
    for (int t = 0; t < 8; ++t)
        acc[t] = __builtin_amdgcn_wmma_f32_16x16x4_f32(
            false, a_cur, false, bf[t], (short)0, acc[t], false, false);

    // Epilogue: add bias, store 16x128 slab.
    const int mrow = m0 + half * 8;
    #pragma unroll
    for (int t = 0; t < 8; ++t) {
        const int n = t * 16 + l15;
        const float bv = bias[n];
        #pragma unroll
        for (int i = 0; i < 8; ++i)
            support[(size_t)(mrow + i) * D_OUT + n] = acc[t][i] + bv;
    }
}

// ---------------------------------------------------------------------------
// Kernel 3: out[row[e]] += vals[e] * support[col[e]]  (COO scatter-add).
// One wave32 per edge; each lane handles 4 consecutive floats (32*4 = 128).
// support & out (51.2 MB each) are L2-resident, so gather + atomics stay on-chip.
// ---------------------------------------------------------------------------
__global__ __launch_bounds__(256) void gc_spmm_scatter(
    const float* __restrict__ support, const float* __restrict__ vals,
    const long long* __restrict__ row, const long long* __restrict__ col,
    float* __restrict__ out, int nnz)
{
    const int e    = (int)((blockIdx.x * 256u + threadIdx.x) >> 5);
    const int lane = threadIdx.x & 31;
    if (e >= nnz) return;

    const int   r = (int)row[e];     // wave-uniform loads
    const int   c = (int)col[e];
    const float v = vals[e];

    const f4 s = *(const f4*)(support + (size_t)c * D_OUT + lane * 4);
    float* o = out + (size_t)r * D_OUT + lane * 4;
    atomicAdd(o + 0, v * s[0]);
    atomicAdd(o + 1, v * s[1]);
    atomicAdd(o + 2, v * s[2]);
    atomicAdd(o + 3, v * s[3]);
}

// ---------------------------------------------------------------------------
extern "C" void kernel_launch(void* const* d_in, const int* in_sizes, int n_in,
                              void* d_out, int out_size, void* d_ws, size_t ws_size,
                              hipStream_t stream) {
    const float*     x    = (const float*)d_in[0];
    const float*     W    = (const float*)d_in[1];
    const float*     b    = (const float*)d_in[2];
    const float*     vals = (const float*)d_in[3];
    const long long* row  = (const long long*)d_in[4];
    const long long* col  = (const long long*)d_in[5];
    float* out     = (float*)d_out;
    float* support = (float*)d_ws;           // M * 128 floats (51.2 MB)

    const int M   = in_sizes[0] / D_IN;      // 100000
    const int nnz = in_sizes[3];             // 1.6M

    // 1) zero the output (harness poisons d_out with 0xAA)
    const int n4 = out_size / 4;
    gc_zero_f4<<<(n4 + 255) / 256, 256, 0, stream>>>((f4*)out, n4);

    // 2) dense projection via fp32 WMMA, W staged in LDS (288 KB dynamic)
    const int gblocks = (M + 127) / 128;     // 128 rows per block (8 waves x 16)
    gc_gemm_wmma_f32<<<gblocks, 256, WLDS_FLOATS * sizeof(float), stream>>>(
        x, W, b, support, M);

    // 3) edge scatter-add (one wave per edge)
    const int sblocks = (nnz + 7) / 8;       // 8 edges per 256-thread block
    gc_spmm_scatter<<<sblocks, 256, 0, stream>>>(support, vals, row, col, out, nnz);
}